// GAT_66855460930109
// MI455X (gfx1250) — compile-verified
//
#include <hip/hip_runtime.h>
#include <hip/hip_bf16.h>
#include <math.h>

// ---------------- problem constants (from reference) ----------------
#define N_NODES 100000
#define N_EDGES 1600000
#define E_TOT   (N_EDGES + N_NODES)   // with self loops
#define F_IN    128
#define HID     8
#define HEADS   8
#define F_MID   (HID * HEADS)         // 64
#define N_CLS   40
#define NEG_SLOPE 0.2f

typedef float v2f __attribute__((ext_vector_type(2)));
typedef float v8f __attribute__((ext_vector_type(8)));

// ---------------- helpers ----------------
__device__ __forceinline__ float leaky(float v) {
    return v > 0.0f ? v : NEG_SLOPE * v;
}
// order-preserving float<->int bijection (involution) for atomicMax on int
__device__ __forceinline__ int enc_f(float f) {
    int i = __float_as_int(f);
    return i >= 0 ? i : (i ^ 0x7fffffff);
}
__device__ __forceinline__ float dec_f(int i) {
    return __int_as_float(i >= 0 ? i : (i ^ 0x7fffffff));
}

// ---------------- init ----------------
__global__ void fill_i32(int* p, int v, int n) {
    int t = blockIdx.x * blockDim.x + threadIdx.x;
    if (t < n) p[t] = v;
}

// Pad + transpose weights: Bt[c][k] = (c < Ncols) ? B[k][c] : 0 ; c in [0, NcolsPad)
__global__ void pad_transpose_B(const float* __restrict__ B, float* __restrict__ Bt,
                                int K, int Ncols, int NcolsPad) {
    int t = blockIdx.x * blockDim.x + threadIdx.x;
    if (t >= K * NcolsPad) return;
    int c = t / K, k = t - c * K;
    Bt[(size_t)c * K + k] = (c < Ncols) ? B[(size_t)k * Ncols + c] : 0.0f;
}

// ---------------- WMMA GEMM: C[N_NODES, Ncols] = A[N_NODES, K] @ B[K, Ncols] ----------------
// One wave per 16-row block; NT column tiles register-blocked per wave.
// Bt ([NT*16][K], zero-padded, transposed) is staged into LDS once per block with the
// CDNA5 async-to-LDS path, then B fragments come from ds_load_b64 (short latency).
// f32 16x16x4 WMMA fragments (ISA 7.12.2, 32-bit layouts):
//   A (16x4):  lanes 0-15 -> M=lane, K offs {0,1}; lanes 16-31 -> M=lane-16, K offs {2,3}
//   B (4x16):  lanes 0-15 -> N=lane, K rows {0,1}; lanes 16-31 -> N=lane-16, K rows {2,3}
//   C/D tile:  VGPR i: lanes 0-15 -> M=i, lanes 16-31 -> M=i+8; N = lane&15
template <int NT, int K>
__global__ void __launch_bounds__(256)
gemm_wmma_f32(const float* __restrict__ A, const float* __restrict__ Bt,
              float* __restrict__ C, int Ncols) {
    __shared__ float sB[NT * 16 * K];                // <= 32 KB

    const int tid  = threadIdx.x;
    const int lane = tid & 31;
    const int wv   = (blockIdx.x * blockDim.x + tid) >> 5;
    const int mTiles = N_NODES / 16;                 // 6250 exactly

    // ---- async copy Bt -> LDS (b128 chunks; chunk count is a multiple of 256) ----
    {
        constexpr int CHUNKS = NT * 16 * K / 4;      // float4 chunks
        const float4* gB = (const float4*)Bt;
        float4*       lB = (float4*)sB;
        for (int i = tid; i < CHUNKS; i += 256) {
            unsigned lds_off = (unsigned)(uintptr_t)(lB + i);  // flat-shared low 32 bits == LDS offset
            const float4* gp = gB + i;
            asm volatile("global_load_async_to_lds_b128 %0, %1, off"
                         :: "v"(lds_off), "v"(gp) : "memory");
        }
        asm volatile("s_wait_asynccnt 0" ::: "memory");
    }
    __syncthreads();                                 // all waves participate (no early exit)

    if (wv < mTiles) {                               // wave-uniform guard
        const int khalf = (lane >> 4) * 2;           // 0 or 2
        const float* arow = A + (size_t)(wv * 16 + (lane & 15)) * K + khalf;
        const int bcol = lane & 15;

        v8f acc[NT];
#pragma unroll
        for (int t = 0; t < NT; ++t) acc[t] = (v8f){0.f,0.f,0.f,0.f,0.f,0.f,0.f,0.f};

#pragma unroll 4
        for (int k = 0; k < K; k += 4) {
            v2f a = *(const v2f*)(arow + k);         // contiguous global b64 load
#pragma unroll
            for (int t = 0; t < NT; ++t) {
                v2f b = *(const v2f*)(&sB[(t * 16 + bcol) * K + k + khalf]);  // ds_load_b64
                acc[t] = __builtin_amdgcn_wmma_f32_16x16x4_f32(
                             /*neg_a=*/false, a, /*neg_b=*/false, b,
                             /*c_mod=*/(short)0, acc[t],
                             /*reuse_a=*/false, /*reuse_b=*/false);
            }
        }

        const int mbase = wv * 16 + ((lane >> 4) << 3);
#pragma unroll
        for (int t = 0; t < NT; ++t) {
            const int col = t * 16 + (lane & 15);
            if (col < Ncols) {
#pragma unroll
                for (int i = 0; i < 8; ++i)
                    C[(size_t)(mbase + i) * Ncols + col] = acc[t][i];
            }
        }
    }
}

// ---------------- attention logit projections: alo[n,h] = sum_c h[n,h,c]*a[h,c] ----------------
__global__ void alo_kernel(const float* __restrict__ h,
                           const float* __restrict__ a_src,
                           const float* __restrict__ a_dst,
                           float* __restrict__ alos, float* __restrict__ alod,
                           int H, int C) {
    int t = blockIdx.x * blockDim.x + threadIdx.x;
    if (t >= N_NODES * H) return;
    int n = t / H, hh = t - n * H;
    const float* hp = h + (size_t)n * H * C + hh * C;
    const float* as = a_src + hh * C;
    const float* ad = a_dst + hh * C;
    float ss = 0.f, sd = 0.f;
    for (int c = 0; c < C; ++c) { float v = hp[c]; ss += v * as[c]; sd += v * ad[c]; }
    alos[t] = ss;
    alod[t] = sd;
}

// ---------------- edge pass A: segment max of leaky logits (ordered-int atomicMax) ----------------
template <int H>
__global__ void edge_max(const int* __restrict__ src, const int* __restrict__ dst,
                         const float* __restrict__ alos, const float* __restrict__ alod,
                         int* __restrict__ m) {
    int t = blockIdx.x * blockDim.x + threadIdx.x;
    if (t >= E_TOT * H) return;
    int e = t / H, hh = t - e * H;
    int s, d;
    if (e < N_EDGES) { s = src[e]; d = dst[e]; } else { s = d = e - N_EDGES; }
    float v = leaky(alos[s * H + hh] + alod[d * H + hh]);
    atomicMax(&m[d * H + hh], enc_f(v));
}

// ---------------- edge pass B, layer 1: exp + denom + 8-wide weighted scatter ----------------
__global__ void edge_acc1(const int* __restrict__ src, const int* __restrict__ dst,
                          const float* __restrict__ alos, const float* __restrict__ alod,
                          const int* __restrict__ m, const float* __restrict__ h1,
                          float* __restrict__ denom, float* __restrict__ acc) {
    int t = blockIdx.x * blockDim.x + threadIdx.x;
    if (t >= E_TOT * HEADS) return;
    int e = t >> 3, hh = t & 7;
    int s, d;
    if (e < N_EDGES) { s = src[e]; d = dst[e]; } else { s = d = e - N_EDGES; }
    float v  = leaky(alos[s * HEADS + hh] + alod[d * HEADS + hh]);
    float mm = dec_f(m[d * HEADS + hh]);
    if (!isfinite(mm)) mm = 0.0f;
    float ex = __expf(v - mm);
    atomicAdd(&denom[d * HEADS + hh], ex);
    const float* hp = h1  + (size_t)s * F_MID + hh * HID;
    float*       ap = acc + (size_t)d * F_MID + hh * HID;
#pragma unroll
    for (int c = 0; c < HID; ++c) atomicAdd(&ap[c], hp[c] * ex);
}

// ---------------- layer-1 normalize: acc/(denom+eps) + b1, ReLU (in place) ----------------
__global__ void norm1(float* __restrict__ acc, const float* __restrict__ denom,
                      const float* __restrict__ b1) {
    int t = blockIdx.x * blockDim.x + threadIdx.x;
    if (t >= N_NODES * F_MID) return;
    int n = t >> 6, hc = t & 63, hh = hc >> 3;
    float dn = denom[n * HEADS + hh] + 1e-16f;
    float v  = acc[t] / dn + b1[hc];
    acc[t] = v > 0.0f ? v : 0.0f;
}

// ---------------- edge pass B, layer 2: 1 head, 40 channels, 8 threads/edge ----------------
__global__ void edge_acc2(const int* __restrict__ src, const int* __restrict__ dst,
                          const float* __restrict__ alos, const float* __restrict__ alod,
                          const int* __restrict__ m, const float* __restrict__ g2,
                          float* __restrict__ denom, float* __restrict__ out) {
    int t = blockIdx.x * blockDim.x + threadIdx.x;
    if (t >= E_TOT * 8) return;
    int e = t >> 3, j = t & 7;
    int s, d;
    if (e < N_EDGES) { s = src[e]; d = dst[e]; } else { s = d = e - N_EDGES; }
    float v  = leaky(alos[s] + alod[d]);
    float mm = dec_f(m[d]);
    if (!isfinite(mm)) mm = 0.0f;
    float ex = __expf(v - mm);
    if (j == 0) atomicAdd(&denom[d], ex);
    const float* gp = g2  + (size_t)s * N_CLS;
    float*       op = out + (size_t)d * N_CLS;
    for (int c = j; c < N_CLS; c += 8) atomicAdd(&op[c], gp[c] * ex);
}

// ---------------- layer-2 normalize: out/(denom+eps) + b2 (in place) ----------------
__global__ void norm2(float* __restrict__ out, const float* __restrict__ denom,
                      const float* __restrict__ b2) {
    int t = blockIdx.x * blockDim.x + threadIdx.x;
    if (t >= N_NODES * N_CLS) return;
    int n = t / N_CLS, c = t - n * N_CLS;
    out[t] = out[t] / (denom[n] + 1e-16f) + b2[c];
}

// ---------------- launcher ----------------
extern "C" void kernel_launch(void* const* d_in, const int* in_sizes, int n_in,
                              void* d_out, int out_size, void* d_ws, size_t ws_size,
                              hipStream_t stream) {
    const float* x     = (const float*)d_in[0];
    const int*   ei    = (const int*)  d_in[1];   // [2, E] flat; src then dst
    const float* W1    = (const float*)d_in[2];
    const float* as1   = (const float*)d_in[3];
    const float* ad1   = (const float*)d_in[4];
    const float* b1    = (const float*)d_in[5];
    const float* W2    = (const float*)d_in[6];
    const float* as2   = (const float*)d_in[7];
    const float* ad2   = (const float*)d_in[8];
    const float* b2    = (const float*)d_in[9];
    float* out = (float*)d_out;

    const int* src = ei;
    const int* dst = ei + N_EDGES;

    // workspace carve-up (256B aligned blocks)
    char* w = (char*)d_ws;
    auto take = [&](size_t bytes) { char* p = w; w += (bytes + 255) & ~size_t(255); return p; };
    float* h1    = (float*)take((size_t)N_NODES * F_MID * 4);
    float* acc1  = (float*)take((size_t)N_NODES * F_MID * 4);   // becomes relu'd hidden
    float* alos1 = (float*)take((size_t)N_NODES * HEADS * 4);
    float* alod1 = (float*)take((size_t)N_NODES * HEADS * 4);
    int*   m1    = (int*)  take((size_t)N_NODES * HEADS * 4);
    float* den1  = (float*)take((size_t)N_NODES * HEADS * 4);
    float* g2    = (float*)take((size_t)N_NODES * N_CLS * 4);
    float* alos2 = (float*)take((size_t)N_NODES * 4);
    float* alod2 = (float*)take((size_t)N_NODES * 4);
    int*   m2    = (int*)  take((size_t)N_NODES * 4);
    float* den2  = (float*)take((size_t)N_NODES * 4);
    float* Bt1   = (float*)take((size_t)F_MID * F_IN * 4);      // [64][128]
    float* Bt2   = (float*)take((size_t)48 * F_MID * 4);        // [48][64] (40 padded to 48)

    const int BS = 256;
    auto nb = [](long n, int bs) { return (int)((n + bs - 1) / bs); };

    // zero accumulators / denominators / output
    hipMemsetAsync(acc1, 0, (size_t)N_NODES * F_MID * 4, stream);
    hipMemsetAsync(den1, 0, (size_t)N_NODES * HEADS * 4, stream);
    hipMemsetAsync(den2, 0, (size_t)N_NODES * 4, stream);
    hipMemsetAsync(out,  0, (size_t)N_NODES * N_CLS * 4, stream);
    // seed segment-max with encode(-inf) ~ INT_MIN
    fill_i32<<<nb((long)N_NODES * HEADS, BS), BS, 0, stream>>>(m1, (int)0x80000000, N_NODES * HEADS);
    fill_i32<<<nb(N_NODES, BS), BS, 0, stream>>>(m2, (int)0x80000000, N_NODES);
    // pad + transpose weights for branch-free, b64-coalesced B fragments
    pad_transpose_B<<<nb((long)F_IN * F_MID, BS), BS, 0, stream>>>(W1, Bt1, F_IN, F_MID, F_MID);
    pad_transpose_B<<<nb((long)F_MID * 48, BS), BS, 0, stream>>>(W2, Bt2, F_MID, N_CLS, 48);

    const int mTiles = N_NODES / 16;   // 6250 waves per GEMM
    const int gemmBlocks = nb((long)mTiles * 32, BS);

    // ---- layer 1 ----
    gemm_wmma_f32<4, F_IN><<<gemmBlocks, BS, 0, stream>>>(x, Bt1, h1, F_MID);
    alo_kernel<<<nb((long)N_NODES * HEADS, BS), BS, 0, stream>>>(h1, as1, ad1, alos1, alod1, HEADS, HID);
    edge_max<HEADS><<<nb((long)E_TOT * HEADS, BS), BS, 0, stream>>>(src, dst, alos1, alod1, m1);
    edge_acc1<<<nb((long)E_TOT * HEADS, BS), BS, 0, stream>>>(src, dst, alos1, alod1, m1, h1, den1, acc1);
    norm1<<<nb((long)N_NODES * F_MID, BS), BS, 0, stream>>>(acc1, den1, b1);

    // ---- layer 2 ----
    gemm_wmma_f32<3, F_MID><<<gemmBlocks, BS, 0, stream>>>(acc1, Bt2, g2, N_CLS);
    alo_kernel<<<nb(N_NODES, BS), BS, 0, stream>>>(g2, as2, ad2, alos2, alod2, 1, N_CLS);
    edge_max<1><<<nb((long)E_TOT, BS), BS, 0, stream>>>(src, dst, alos2, alod2, m2);
    edge_acc2<<<nb((long)E_TOT * 8, BS), BS, 0, stream>>>(src, dst, alos2, alod2, m2, g2, den2, out);
    norm2<<<nb((long)N_NODES * N_CLS, BS), BS, 0, stream>>>(out, den2, b2);
}